// MatrixTree_83915071029664
// MI455X (gfx1250) — compile-verified
//
#include <hip/hip_runtime.h>
#include <hip/hip_bf16.h>
#include <math.h>

// Matrix-Tree marginals:  probs = d logZ / d scores
//   probs[i][j] = A[i][j] * (Minv[i][i] - Minv[j][i])   (i>=1, pair; else 0)
// Identity rows (unmasked tokens + padded root row/col) decouple, so we invert
// only the compacted masked submatrix (na = #masked tokens, padded to 16) with
// blocked Gauss-Jordan driven by V_WMMA_F32_16X16X4_F32 rank-16 updates.
// One workgroup (8 wave32) per batch; matrix lives entirely in LDS (320KB WGP).

#define SDIM 256
#define NSQ (SDIM * SDIM)
#define MSTRIDE (SDIM + 1) // odd stride -> bank-conflict-free column access
#define CSTRIDE 17

typedef float v2f __attribute__((ext_vector_type(2)));
typedef float v8f __attribute__((ext_vector_type(8)));

// A-fragment of a 16x16 tile for V_WMMA_F32_16X16X4_F32 (4 chained K=4 chunks).
// Lane m=lane&15 holds row m; half=lane>>4 selects the K pair within a chunk.
struct AFrag { v2f a[4]; };

__device__ __forceinline__ AFrag load_afrag(const float* A, int as, bool neg, int lane)
{
    const int m = lane & 15, h = lane >> 4;
    AFrag f;
#pragma unroll
    for (int kq = 0; kq < 4; ++kq) {
        const int kb = kq * 4 + 2 * h;
        float x = A[m * as + kb];
        float y = A[m * as + kb + 1];
        if (neg) { x = -x; y = -y; }
        f.a[kq].x = x; f.a[kq].y = y;
    }
    return f;
}

// acc += Afrag * B   (16x16x16 via 4 chained V_WMMA_F32_16X16X4_F32)
__device__ __forceinline__ v8f wmma_apply(const AFrag& f, const float* B, int bs,
                                          v8f acc, int lane)
{
    const int n = lane & 15, h = lane >> 4;
#pragma unroll
    for (int kq = 0; kq < 4; ++kq) {
        const int kb = kq * 4 + 2 * h;
        v2f b;
        b.x = B[kb * bs + n];
        b.y = B[(kb + 1) * bs + n];
        acc = __builtin_amdgcn_wmma_f32_16x16x4_f32(false, f.a[kq], false, b,
                                                    (short)0, acc, false, false);
    }
    return acc;
}

// C/D 16x16 f32 layout: VGPR v, lanes 0-15 -> row v, lanes 16-31 -> row v+8.
__device__ __forceinline__ v8f load_tile(const float* C, int cs, int lane)
{
    const int n = lane & 15, mb = (lane >> 4) * 8;
    v8f c;
#pragma unroll
    for (int v = 0; v < 8; ++v) c[v] = C[(mb + v) * cs + n];
    return c;
}

__device__ __forceinline__ void store_tile(float* D, int dstr, v8f d, int lane)
{
    const int n = lane & 15, mb = (lane >> 4) * 8;
#pragma unroll
    for (int v = 0; v < 8; ++v) D[(mb + v) * dstr + n] = d[v];
}

__global__ __launch_bounds__(256, 1)
void mtt_marginals_kernel(const float* __restrict__ scores,
                          const unsigned char* __restrict__ mask, // jnp bool: 1B
                          float* __restrict__ out)
{
    __shared__ float Msh[SDIM][MSTRIDE];  // 263,168 B  compact matrix / inverse
    __shared__ float Csh[SDIM][CSTRIDE];  //  17,408 B  column-panel save
    __shared__ float Pinv[16][17];        //   1,088 B  pivot-block inverse
    __shared__ float red[SDIM];           //   1,024 B
    __shared__ float c0[SDIM];            //   1,024 B  root-column weights A[i][0]
    __shared__ int   mr[SDIM];            //   1,024 B  mask_r
    __shared__ int   idxs[SDIM];          //   1,024 B  compact -> original row
    __shared__ int   ra[SDIM];            //   1,024 B  original -> compact row
    __shared__ int   na_sh;
    __shared__ float msh;

    const int b    = blockIdx.x;
    const int t    = threadIdx.x;
    const int lane = t & 31;
    const int wave = t >> 5;
    const float* sb = scores + (size_t)b * NSQ;
    float*       ob = out    + (size_t)b * NSQ;

    mr[t] = (t == 0) ? 1 : (mask[(size_t)b * SDIM + t] != 0);
    __syncthreads();

    if (t == 0) { // active set = masked token rows (i>=1, mr[i]); root row excluded
        int n = 0;
        ra[0] = -1;
        for (int i = 1; i < SDIM; ++i) {
            ra[i] = -1;
            if (mr[i]) { idxs[n] = i; ra[i] = n; ++n; }
        }
        na_sh = n;
    }

    // ---- phase 0: m = max over pair-masked scores (coalesced full scan)
    const int cj = mr[t];
    float lm = -INFINITY;
    for (int k = 0; k < SDIM; ++k) {
        float s = sb[k * SDIM + t];
        if (cj && mr[k]) lm = fmaxf(lm, s);
    }
    red[t] = lm;
    __syncthreads();
    for (int off = 128; off > 0; off >>= 1) {
        if (t < off) red[t] = fmaxf(red[t], red[t + off]);
        __syncthreads();
    }
    if (t == 0) msh = red[0];
    __syncthreads();
    const float mval = msh;
    const int   na   = na_sh;
    const int   nac  = (na + 15) & ~15;   // padded active size
    const int   nbb  = nac >> 4;          // number of 16-blocks

    // ---- phase 1a: compact A into Msh; root-column weights into c0.
    // Mask is a contiguous range in practice, so these loads remain coalesced.
    {
        const int myIdx = (t < na) ? idxs[t] : -1;
        if (t < na) {
            c0[t] = expf(sb[(size_t)myIdx * SDIM] - mval); // A[idx_t][0]
            for (int a = 0; a < na; ++a)
                Msh[a][t] = expf(sb[(size_t)idxs[a] * SDIM + myIdx] - mval);
        }
    }
    __syncthreads();

    // ---- phase 1b: compact Laplacian; diag = A[i][0] + sum_offdiag (strictly d.d.)
    if (t < na) {
        float D = c0[t];
        for (int bcol = 0; bcol < na; ++bcol) D += Msh[t][bcol];
        const float Aii = Msh[t][t];
        for (int bcol = 0; bcol < nac; ++bcol) {
            float a = (bcol < na) ? Msh[t][bcol] : 0.0f;
            Msh[t][bcol] = (bcol == t) ? (D - Aii) : -a;
        }
    } else if (t < nac) {
        for (int bcol = 0; bcol < nac; ++bcol) Msh[t][bcol] = (bcol == t) ? 1.0f : 0.0f;
    }
    __syncthreads();

    // ---- phase 2: blocked Gauss-Jordan inversion in place (pivot-free, d.d.)
    for (int kstep = 0; kstep < nbb; ++kstep) {
        const int kb = kstep * 16;

        if (wave == 0) {
            if (lane < 16) { // invert 16x16 pivot block with shuffle GJ
                const int r = lane;
                float p[16], q[16];
#pragma unroll
                for (int j = 0; j < 16; ++j) {
                    p[j] = Msh[kb + r][kb + j];
                    q[j] = (j == r) ? 1.0f : 0.0f;
                }
#pragma unroll
                for (int c = 0; c < 16; ++c) {
                    float piv = __shfl(p[c], c, 32);
                    float inv = 1.0f / piv;
                    if (r == c) {
#pragma unroll
                        for (int j = 0; j < 16; ++j) { p[j] *= inv; q[j] *= inv; }
                    }
                    float f = (r == c) ? 0.0f : p[c];
#pragma unroll
                    for (int j = 0; j < 16; ++j) {
                        float pc = __shfl(p[j], c, 32);
                        float qc = __shfl(q[j], c, 32);
                        p[j] = fmaf(-f, pc, p[j]);
                        q[j] = fmaf(-f, qc, q[j]);
                    }
                }
#pragma unroll
                for (int j = 0; j < 16; ++j) Pinv[r][j] = q[j];
            }
        } else {
            // waves 1-7: save column panel (non-pivot rows x 16) and zero it
            const int t2   = t - 32;            // 0..223
            const int rows = nac - 16;
            for (int e = t2; e < rows * 16; e += 224) {
                int rr = e >> 4;
                int c  = e & 15;
                int i  = rr + ((rr >= kb) ? 16 : 0);
                float v = Msh[i][kb + c];
                Csh[i][c] = v;
                Msh[i][kb + c] = 0.0f;
            }
        }
        __syncthreads();

        // row-panel: M[k][j] <- Pinv * M[k][j] (j != k); Pinv fragment stays resident
        {
            AFrag pf = load_afrag(&Pinv[0][0], 17, false, lane);
            for (int tt = wave; tt < nbb - 1; tt += 8) {
                int bj = tt + ((tt >= kstep) ? 1 : 0);
                v8f acc = {};
                acc = wmma_apply(pf, &Msh[kb][bj * 16], MSTRIDE, acc, lane);
                store_tile(&Msh[kb][bj * 16], MSTRIDE, acc, lane);
            }
        }
        Msh[kb + (t >> 4)][kb + (t & 15)] = Pinv[t >> 4][t & 15]; // M[k][k] <- Pinv
        __syncthreads();

        // rank-16 update: M[i][j] -= C[i]*M[k][j]; -C fragment resident per row-block
        for (int r_ = wave; r_ < nbb - 1; r_ += 8) {
            int bi = r_ + ((r_ >= kstep) ? 1 : 0);
            AFrag cf = load_afrag(&Csh[bi * 16][0], CSTRIDE, /*neg=*/true, lane);
            for (int bj = 0; bj < nbb; ++bj) {
                float* tile = &Msh[bi * 16][bj * 16];
                v8f acc = load_tile(tile, MSTRIDE, lane);
                acc = wmma_apply(cf, &Msh[kb][bj * 16], MSTRIDE, acc, lane);
                store_tile(tile, MSTRIDE, acc, lane);
            }
        }
        __syncthreads();
    }
    // Msh now holds the compact inverse.

    // ---- phase 3: probs[i][j] = A[i][j]*(Minv[i][i] - Minv[j][i]); Minv[0][i]=0.
    {
        const int rj = (t >= 1 && cj) ? ra[t] : -1; // compact column row for j=t
        for (int k = 0; k < SDIM; ++k) {
            float v = 0.0f;
            const int ri = ra[k]; // valid iff k>=1 && mr[k]
            if (ri >= 0 && (t == 0 || rj >= 0)) {
                float s   = sb[k * SDIM + t];
                float dia = Msh[ri][ri];                    // LDS broadcast
                float mji = (t == 0) ? 0.0f : Msh[rj][ri];  // conflict-free column
                v = expf(s - mval) * (dia - mji);
            }
            ob[k * SDIM + t] = v;
        }
    }
}

extern "C" void kernel_launch(void* const* d_in, const int* in_sizes, int n_in,
                              void* d_out, int out_size, void* d_ws, size_t ws_size,
                              hipStream_t stream)
{
    (void)n_in; (void)out_size; (void)d_ws; (void)ws_size;
    const float*         scores = (const float*)d_in[0];
    const unsigned char* mask   = (const unsigned char*)d_in[1]; // bool[B,S]
    float*               out    = (float*)d_out;
    const int B = in_sizes[1] / SDIM; // 256
    hipLaunchKernelGGL(mtt_marginals_kernel, dim3(B), dim3(256), 0, stream,
                       scores, mask, out);
}